// SpikeFP64Sqrt_43860206027323
// MI455X (gfx1250) — compile-verified
//
#include <hip/hip_runtime.h>

// SpikeFP64Sqrt for MI455X (gfx1250, wave32).
// One wave32 handles 32 consecutive fp64 elements (each = 64 float32 bit-pulses):
//   pack:   64 coalesced NT b32 loads (clause-batched) -> ballot_w32 + v_bfrev
//   math:   exponent-halving guess + 12 bit-exact IEEE fp64 Newton iterations
//   unpack: v_readlane broadcast -> per-lane bit-pair extract -> 32 NT b64 stores
// All control flow is scalar (readfirstlane-uniform wave index + full/tail split),
// so no per-iteration EXEC-mask churn. 512 MB stream -> ~22us HBM floor.

typedef float vf2 __attribute__((ext_vector_type(2)));

#define SQRT2_MANT 1865452045155277ULL
#define NAN_BITS   0x7FF8000000000000ULL
#define INF_BITS   0x7FF0000000000000ULL
#define MANT_MASK  0x000FFFFFFFFFFFFFULL

__global__ __launch_bounds__(256) void spike_fp64_sqrt_kernel(
    const float* __restrict__ x, float* __restrict__ out, int n) {
  const int lane = threadIdx.x & 31;
  // Wave index is wave-uniform: pin it to an SGPR so all guards are scalar.
  const int wib = __builtin_amdgcn_readfirstlane((int)(threadIdx.x >> 5));
  const long long wid = (long long)blockIdx.x * 8 + wib;
  const long long ebase = wid << 5;  // first of this wave's 32 elements
  if (ebase >= n) return;           // scalar branch

  const bool full_tile = (ebase + 32) <= (long long)n;  // scalar
  const float* __restrict__ xg = x + ebase * 64;

  // ---------------- pack: wave-cooperative transpose via ballot ----------------
  unsigned int my_hi = 0u, my_lo = 0u;
  if (full_tile) {
    float p0[32], p1[32];
#pragma unroll
    for (int j = 0; j < 32; ++j) {  // pure load phase: clause-friendly NT stream
      p0[j] = __builtin_nontemporal_load(xg + j * 64 + lane);       // bits 63..32
      p1[j] = __builtin_nontemporal_load(xg + j * 64 + 32 + lane);  // bits 31..0
    }
#pragma unroll
    for (int j = 0; j < 32; ++j) {
      unsigned int hi = __brev(__builtin_amdgcn_ballot_w32(p0[j] > 0.5f));
      unsigned int lo = __brev(__builtin_amdgcn_ballot_w32(p1[j] > 0.5f));
      if (lane == j) { my_hi = hi; my_lo = lo; }
    }
  } else {
#pragma unroll 1
    for (int j = 0; j < 32; ++j) {
      if (ebase + j < n) {  // scalar guard (ebase, n both SGPR)
        float p0 = __builtin_nontemporal_load(xg + j * 64 + lane);
        float p1 = __builtin_nontemporal_load(xg + j * 64 + 32 + lane);
        unsigned int hi = __brev(__builtin_amdgcn_ballot_w32(p0 > 0.5f));
        unsigned int lo = __brev(__builtin_amdgcn_ballot_w32(p1 > 0.5f));
        if (lane == j) { my_hi = hi; my_lo = lo; }
      }
    }
  }

  const unsigned long long u = ((unsigned long long)my_hi << 32) | my_lo;
  const double f = __longlong_as_double((long long)u);

  // ---- sqrt guess: y0 = 2^floor(E/2) * (sqrt2 mantissa if E odd else 1.0) ----
  const int e_x = (int)((u >> 52) & 0x7FFULL);
  const int e_real = e_x - 1023;
  const bool is_odd = (e_real & 1) == 1;                    // two's-complement parity
  const long long e_new = (long long)(e_real >> 1) + 1023;  // arithmetic shift = floor/2
  const unsigned long long mant = is_odd ? SQRT2_MANT : 0ULL;
  const unsigned long long ybits = (((unsigned long long)e_new) << 52) | mant;
  double y = __longlong_as_double((long long)ybits);

  // ---- 12 Newton-Raphson iterations, IEEE fp64 (RTE, correct division) ----
#pragma unroll
  for (int it = 0; it < 12; ++it) {
    y = 0.5 * (y + f / y);
  }

  // ---- special-case muxes, same order as reference ----
  unsigned long long r = (unsigned long long)__double_as_longlong(y);
  const bool is_neg    = (u >> 63) != 0ULL;
  const bool m_any     = (u & MANT_MASK) != 0ULL;
  const bool e_all_one = (e_x == 0x7FF);
  const bool e_is_zero = (e_x == 0);
  if (is_neg)                         r = NAN_BITS;
  if (e_all_one && m_any)             r = NAN_BITS;
  if (e_all_one && !m_any && !is_neg) r = INF_BITS;
  if (e_is_zero && !m_any)            r = 0ULL;

  const int r_hi = (int)(r >> 32);
  const int r_lo = (int)(r & 0xFFFFFFFFULL);

  // ---------------- unpack: readlane broadcast + bit-pair b64 stores ----------------
  float* __restrict__ og = out + ebase * 64;
  if (full_tile) {
    const int s0 = 31 - 2 * (lane & 15);  // shift for first bit of this lane's pair
#pragma unroll
    for (int j = 0; j < 32; ++j) {
      unsigned int hj = (unsigned int)__builtin_amdgcn_readlane(r_hi, j);
      unsigned int lj = (unsigned int)__builtin_amdgcn_readlane(r_lo, j);
      unsigned int w = (lane < 16) ? hj : lj;  // lanes 0..15 -> hi word, 16..31 -> lo
      vf2 o;
      o.x = (float)((w >> s0) & 1u);        // pulse k = 2*lane
      o.y = (float)((w >> (s0 - 1)) & 1u);  // pulse k = 2*lane + 1
      __builtin_nontemporal_store(o, (vf2*)(og + j * 64 + 2 * lane));
    }
  } else {
#pragma unroll 1
    for (int j = 0; j < 32; ++j) {
      if (ebase + j < n) {  // scalar guard
        unsigned int hj = (unsigned int)__builtin_amdgcn_readlane(r_hi, j);
        unsigned int lj = (unsigned int)__builtin_amdgcn_readlane(r_lo, j);
        float o0 = (float)((hj >> (31 - lane)) & 1u);
        float o1 = (float)((lj >> (31 - lane)) & 1u);
        __builtin_nontemporal_store(o0, og + j * 64 + lane);
        __builtin_nontemporal_store(o1, og + j * 64 + 32 + lane);
      }
    }
  }
}

extern "C" void kernel_launch(void* const* d_in, const int* in_sizes, int n_in,
                              void* d_out, int out_size, void* d_ws, size_t ws_size,
                              hipStream_t stream) {
  const float* x = (const float*)d_in[0];
  float* out = (float*)d_out;

  const long long n = (long long)in_sizes[0] / 64;  // number of fp64 elements
  const int waves_needed = (int)((n + 31) / 32);
  const int waves_per_block = 8;  // 256 threads = 8 wave32
  const int blocks = (waves_needed + waves_per_block - 1) / waves_per_block;

  spike_fp64_sqrt_kernel<<<blocks, 256, 0, stream>>>(x, out, (int)n);
}